// LowPassFilter_78786880078251
// MI455X (gfx1250) — compile-verified
//
#include <hip/hip_runtime.h>

typedef __attribute__((ext_vector_type(2))) float v2f;
typedef __attribute__((ext_vector_type(8))) float v8f;

__device__ __forceinline__ float wsel(int j, float g0, float g1, float g2, float g3, float g4)
{
    float r = 0.f;
    r = (j == 0) ? g0 : r;
    r = (j == 1) ? g1 : r;
    r = (j == 2) ? g2 : r;
    r = (j == 3) ? g3 : r;
    r = (j == 4) ? g4 : r;
    return r;
}

// Build the per-(axis_tile, k_block, lane) banded A-matrix table for
// V_WMMA_F32_16X16X4_F32.  A[m][k] = g[4b + k - m] (band, else 0), with the
// zero-padding at the volume borders folded in by zeroing A columns whose
// input row zb-2+4b+k falls outside [0,63].  g[] is recovered from the
// normalized 3D kernel by row sums (k3 = g (x) g (x) g, sum(k3)=1).
// Table layout: Atab[((a*5 + b)*32 + lane)*2 + {x,y}]  (4*5*32*2 floats).
__global__ void lp_prep(const float* __restrict__ k3, float* __restrict__ Atab)
{
    const int t = blockIdx.x * blockDim.x + threadIdx.x;
    if (t >= 4 * 5 * 32) return;

    float gs[5];
    #pragma unroll
    for (int i = 0; i < 5; ++i) {
        float s = 0.f;
        #pragma unroll
        for (int j = 0; j < 25; ++j) s += k3[i * 25 + j];
        gs[i] = s;
    }

    const int lane = t & 31;
    const int b    = (t >> 5) % 5;
    const int a    = t / (5 * 32);
    const int mm   = lane & 15;
    const int half = lane >> 4;
    const int zb   = a * 16;
    const int k0   = 4 * b + 2 * half;

    float ax = wsel(k0 - mm,     gs[0], gs[1], gs[2], gs[3], gs[4]);
    float ay = wsel(k0 + 1 - mm, gs[0], gs[1], gs[2], gs[3], gs[4]);
    const int z0 = zb - 2 + k0;           // input row fed by A column k0
    if ((unsigned)z0       >= 64u) ax = 0.f;
    if ((unsigned)(z0 + 1) >= 64u) ay = 0.f;

    Atab[t * 2 + 0] = ax;
    Atab[t * 2 + 1] = ay;
}

// One wave32 computes a 16(axis) x 16(channel) f32 output tile:
//   Out[m][n] = sum_{j=0..4} g[j] * In[zb + m + j - 2][c0 + n]
// as 5 chained V_WMMA_F32_16X16X4_F32.  A tiles come from the precomputed
// table; B loads are clamped to [0,63] (border rows are multiplied by the
// zeroed A columns), so the whole kernel is branch-free with EXEC all-1s.
__global__ __launch_bounds__(256) void lp_axis_pass(
    const float* __restrict__ in, float* __restrict__ out,
    const float* __restrict__ Atab, int S, int m1, int m2)
{
    const int wid  = (blockIdx.x * blockDim.x + threadIdx.x) >> 5;
    const int lane = threadIdx.x & 31;
    const int mm   = lane & 15;   // N (B/output column) index
    const int half = lane >> 4;   // K-half for A/B, M+8 selector for C/D

    const int c_tile = wid & 1;        // 32 channels -> 2 tiles of 16
    const int a_tile = (wid >> 1) & 3; // 64 axis positions -> 4 tiles of 16
    const int o      = wid >> 3;       // 4*64*64 = 16384 other-dim combos
    const int i2 = o & 63;
    const int i1 = (o >> 6) & 63;
    const int i0 = o >> 12;
    const unsigned base = (unsigned)i0 * 8388608u
                        + (unsigned)i1 * (unsigned)m1
                        + (unsigned)i2 * (unsigned)m2;

    const int zb = a_tile * 16;
    const int cc = c_tile * 16 + mm;
    const float* p = in + (base + (unsigned)cc);
    const v2f*  At = (const v2f*)Atab + (a_tile * 5) * 32 + lane;

    v8f acc = {};
    #pragma unroll
    for (int b = 0; b < 5; ++b) {
        const v2f A = At[b * 32];                  // coalesced 8B/lane
        const int z0 = zb - 2 + 4 * b + 2 * half;  // input rows for B.x/B.y
        const int za = min(max(z0, 0), 63);        // clamped: always in-bounds
        const int zc = min(max(z0 + 1, 0), 63);
        v2f B;
        B.x = p[(unsigned)(za * S)];
        B.y = p[(unsigned)(zc * S)];
        acc = __builtin_amdgcn_wmma_f32_16x16x4_f32(
            /*neg_a=*/false, A, /*neg_b=*/false, B,
            /*c_mod=*/(short)0, acc, /*reuse_a=*/false, /*reuse_b=*/false);
    }

    float* q = out + (base + (unsigned)cc);
    #pragma unroll
    for (int r = 0; r < 8; ++r) {
        const int zo = zb + r + 8 * half;          // C/D layout: M = r + 8*half
        q[(unsigned)(zo * S)] = acc[r];
    }
}

extern "C" void kernel_launch(void* const* d_in, const int* in_sizes, int n_in,
                              void* d_out, int out_size, void* d_ws, size_t ws_size,
                              hipStream_t stream)
{
    (void)in_sizes; (void)n_in; (void)out_size; (void)ws_size;
    const float* img  = (const float*)d_in[0];  // [4,64,64,64,32] f32
    const float* k3   = (const float*)d_in[1];  // [1,1,5,5,5] f32
    float* out  = (float*)d_out;
    float* Atab = (float*)d_ws;                         // 4*5*32*2 floats (5 KB)
    float* tmp  = (float*)((char*)d_ws + 8192);         // 128 MiB ping buffer

    lp_prep<<<5, 128, 0, stream>>>(k3, Atab);

    const dim3 grid(16384), block(256);  // 131072 wave32 tiles per pass
    // Z pass (innermost spatial): img -> out.  S=32;     base over (t, x, y)
    lp_axis_pass<<<grid, block, 0, stream>>>(img, out, Atab, 32,     131072, 2048);
    // Y pass: out -> tmp.                      S=2048;   base over (t, x, z)
    lp_axis_pass<<<grid, block, 0, stream>>>(out, tmp, Atab, 2048,   131072, 32);
    // X pass: tmp -> out.                      S=131072; base over (t, y, z)
    lp_axis_pass<<<grid, block, 0, stream>>>(tmp, out, Atab, 131072, 2048,   32);
}